// AdaptiveDiscretizedNeuralODE8_Spatial_pointwise_view_17_89395449299422
// MI455X (gfx1250) — compile-verified
//
#include <hip/hip_runtime.h>

typedef float f2  __attribute__((ext_vector_type(2)));
typedef float f4  __attribute__((ext_vector_type(4)));
typedef float v2f __attribute__((ext_vector_type(2)));
typedef float v8f __attribute__((ext_vector_type(8)));

#define NLAYER 30
#define CCH    256   // channels = 16*16
#define HW     1024  // 32*32
#define NBAT   16
#define NROW   4096  // NBAT*CCH, reduction length per p

// Sum a value held by all 32 lanes of the wave, using v_wmma_f32_16x16x4_f32
// against a ones matrix.  Stage 1: A=(q,0) -> D[m,*] = q[m] + q[m+16]
// (A layout: lanes 0-15 supply K0/K1, lanes 16-31 supply K2/K3).
// Stage 2: feed D vgpr pairs back as A with C-accumulation; after 4 WMMAs every
// lane/vgpr of acc holds sum over all 32 lanes.  Requires EXEC == all ones.
__device__ __forceinline__ float wmma_sum32(float q) {
  v2f a;   a[0] = q;   a[1] = 0.0f;
  v2f one; one[0] = 1.0f; one[1] = 1.0f;
  v8f c0 = {0.f, 0.f, 0.f, 0.f, 0.f, 0.f, 0.f, 0.f};
  v8f d = __builtin_amdgcn_wmma_f32_16x16x4_f32(false, a, false, one, (short)0, c0, false, false);
  v8f acc = c0;
  v2f p;
  p[0] = d[0]; p[1] = d[1];
  acc = __builtin_amdgcn_wmma_f32_16x16x4_f32(false, p, false, one, (short)0, acc, false, false);
  p[0] = d[2]; p[1] = d[3];
  acc = __builtin_amdgcn_wmma_f32_16x16x4_f32(false, p, false, one, (short)0, acc, false, false);
  p[0] = d[4]; p[1] = d[5];
  acc = __builtin_amdgcn_wmma_f32_16x16x4_f32(false, p, false, one, (short)0, acc, false, false);
  p[0] = d[6]; p[1] = d[7];
  acc = __builtin_amdgcn_wmma_f32_16x16x4_f32(false, p, false, one, (short)0, acc, false, false);
  return acc[0];
}

__global__ __launch_bounds__(512) void ode_fused(
    const float* __restrict__ x,     // [16,256,32,32]
    const float* __restrict__ dtp,   // [30]
    const float* __restrict__ mats,  // [30,256]
    const float* __restrict__ gam,   // [1024]
    const float* __restrict__ bet,   // [1024]
    float* __restrict__ out)         // [16,256,32,32]
{
  __shared__ float mtab[NLAYER * CCH];  // 30 KB: m for all layers
  __shared__ f4    red[2][32];          // double-buffered wave partials {s1.x,s1.y,s2.x,s2.y}

  const int tid  = threadIdx.x;
  const int lane = tid & 31;
  const int wave = tid >> 5;
  const int p0   = blockIdx.x * 2;   // this block owns p0, p0+1
  const int c    = tid & 255;        // all 8 rows of this thread share channel c

  // ---- stage m table into LDS via async global->LDS (ASYNCcnt path) ----
  for (int j = tid; j < (NLAYER * CCH) / 4; j += 512) {
    const float* g = mats + 4 * j;
    unsigned loff = (unsigned)(unsigned long long)&mtab[4 * j];
    asm volatile("global_load_async_to_lds_b128 %0, %1, off"
                 :: "v"(loff), "v"(g) : "memory");
  }
  // zero both parities of red; only slots 0..15 get rewritten each layer,
  // slots 16..31 must stay 0 so the 32-lane WMMA reduction is exact.
  if (tid < 64) {
    f4 z; z.x = 0.f; z.y = 0.f; z.z = 0.f; z.w = 0.f;
    red[tid >> 5][tid & 31] = z;
  }

  // ---- load x tile into registers; y0 := x1 ----
  f2 X[8], Y[8];
#pragma unroll
  for (int k = 0; k < 8; ++k) {
    const size_t r = (size_t)(tid + 512 * k);   // r = b*256 + c
    X[k] = *(const f2*)(x + r * HW + p0);
    Y[k] = X[k];
  }
  f2 g2 = *(const f2*)(gam + p0);
  f2 b2 = *(const f2*)(bet + p0);

  asm volatile("s_wait_asynccnt 0x0" ::: "memory");
  __syncthreads();

  const float inv = 1.0f / (float)NROW;
  const f2 zero2 = {0.f, 0.f};
  const f2 six2  = {6.f, 6.f};

#pragma unroll 1
  for (int l = 0; l < NLAYER; ++l) {
    const float mc = mtab[l * CCH + c];
    float dt = dtp[l];
    dt = fminf(fmaxf(dt, 0.0f), 6.0f);

    f2 Z[8];
    f2 s1 = zero2, s2 = zero2;
#pragma unroll
    for (int k = 0; k < 8; ++k) {
      f2 z = Y[k] + mc * X[k];
      Z[k] = z;
      s1 += z;
      s2 += z * z;
    }

    // intra-wave butterfly: 32 lanes -> wave partial
#pragma unroll
    for (int m = 16; m > 0; m >>= 1) {
      s1.x += __shfl_xor(s1.x, m, 32);
      s1.y += __shfl_xor(s1.y, m, 32);
      s2.x += __shfl_xor(s2.x, m, 32);
      s2.y += __shfl_xor(s2.y, m, 32);
    }
    const int par = l & 1;
    if (lane == 0) {
      f4 v; v.x = s1.x; v.y = s1.y; v.z = s2.x; v.w = s2.y;
      red[par][wave] = v;
    }
    __syncthreads();

    // cross-wave reduction: every (full) wave folds 32 partial slots via WMMA
    f4 q = red[par][lane];
    float t0 = wmma_sum32(q.x);  // sum z  for p0
    float t1 = wmma_sum32(q.y);  // sum z  for p0+1
    float t2 = wmma_sum32(q.z);  // sum z^2 for p0
    float t3 = wmma_sum32(q.w);  // sum z^2 for p0+1

    f2 mean; mean.x = t0 * inv;                  mean.y = t1 * inv;
    f2 var;  var.x  = t2 * inv - mean.x * mean.x; var.y = t3 * inv - mean.y * mean.y;
    f2 sc;
    sc.x = rsqrtf(var.x + 1e-5f) * g2.x;
    sc.y = rsqrtf(var.y + 1e-5f) * g2.y;
    f2 bi = b2 - mean * sc;

#pragma unroll
    for (int k = 0; k < 8; ++k) {
      f2 a = Z[k] * sc + bi;
      a = __builtin_elementwise_max(a, zero2);
      a = __builtin_elementwise_min(a, six2);
      Y[k] += dt * (a - Y[k]);     // y0 + dt*(-y0 + act)
    }
  }

  // ---- epilogue: out = y0 + x1, same flat layout as input ----
#pragma unroll
  for (int k = 0; k < 8; ++k) {
    const size_t r = (size_t)(tid + 512 * k);
    f2 o = Y[k] + X[k];
    *(f2*)(out + r * HW + p0) = o;
  }
}

extern "C" void kernel_launch(void* const* d_in, const int* in_sizes, int n_in,
                              void* d_out, int out_size, void* d_ws, size_t ws_size,
                              hipStream_t stream) {
  const float* x    = (const float*)d_in[0];
  const float* dtp  = (const float*)d_in[1];
  const float* mats = (const float*)d_in[2];
  const float* gam  = (const float*)d_in[3];
  const float* bet  = (const float*)d_in[4];
  float* out = (float*)d_out;
  // 512 blocks x 2 p-values = 1024 spatial positions; 512 threads = 16 waves.
  ode_fused<<<512, 512, 0, stream>>>(x, dtp, mats, gam, bet, out);
}